// EVD_30631706755680
// MI455X (gfx1250) — compile-verified
//
#include <hip/hip_runtime.h>
#include <hip/hip_bf16.h>
#include <math.h>

typedef __attribute__((ext_vector_type(2))) float v2f;
typedef __attribute__((ext_vector_type(8))) float v8f;

#define BATCH 256
#define CH    64
#define NN    4096

// ---------------------------------------------------------------------------
// Kernel 1: per-batch covariance  S[b] = y[b] @ y[b]^T / N   (64x64 out, K=4096)
// One block per batch, 8 waves. Wave w: A row-block mi = w>>1, B row-blocks
// {2*(w&1), 2*(w&1)+1}. fp32 WMMA 16x16x4.
// ---------------------------------------------------------------------------
__global__ __launch_bounds__(256) void cov_kernel(const float* __restrict__ y,
                                                  float* __restrict__ S) {
    const int b    = blockIdx.x;
    const int wave = threadIdx.x >> 5;
    const int lane = threadIdx.x & 31;
    const int m    = lane & 15;   // row within block (A) / col within tile (B)
    const int h    = lane >> 4;   // K-half select

    const int mi  = wave >> 1;
    const int nj  = wave & 1;
    const int ni0 = nj * 2;
    const int ni1 = nj * 2 + 1;

    const float* yb   = y + (size_t)b * CH * NN;
    const float* rowA = yb + (size_t)(mi  * 16 + m) * NN + 2 * h;
    const float* rowB0 = yb + (size_t)(ni0 * 16 + m) * NN + 2 * h;
    const float* rowB1 = yb + (size_t)(ni1 * 16 + m) * NN + 2 * h;

    v8f acc0 = {};
    v8f acc1 = {};

    #pragma unroll 4
    for (int kk = 0; kk < NN; kk += 4) {
        v2f a  = *(const v2f*)(rowA  + kk);
        v2f b0 = *(const v2f*)(rowB0 + kk);
        v2f b1 = *(const v2f*)(rowB1 + kk);
        acc0 = __builtin_amdgcn_wmma_f32_16x16x4_f32(false, a, false, b0,
                                                     (short)0, acc0, false, false);
        acc1 = __builtin_amdgcn_wmma_f32_16x16x4_f32(false, a, false, b1,
                                                     (short)0, acc1, false, false);
    }

    const float scale = 1.0f / (float)NN;
    float* Sb = S + (size_t)b * CH * CH;
    const int col0 = ni0 * 16 + m;
    const int col1 = ni1 * 16 + m;
    #pragma unroll
    for (int r = 0; r < 8; ++r) {
        const int row = mi * 16 + r + 8 * h;
        Sb[row * CH + col0] = acc0[r] * scale;
        Sb[row * CH + col1] = acc1[r] * scale;
    }
}

// ---------------------------------------------------------------------------
// Kernel 2: parallel cyclic Jacobi eigensolver for 64x64 symmetric S.
// One block (64 threads = 2 waves) per batch. 32 disjoint pairs per round
// (round-robin tournament), 63 rounds/sweep, 10 sweeps. U accumulated in LDS.
// Eigen-columns written sorted by ascending eigenvalue (eigh convention).
// ---------------------------------------------------------------------------
__global__ __launch_bounds__(64) void jacobi_kernel(const float* __restrict__ S_in,
                                                    float* __restrict__ U_out) {
    __shared__ float S[CH][CH + 1];
    __shared__ float U[CH][CH + 1];
    __shared__ float rc[32], rs[32];
    __shared__ int   pp[32], qq[32];
    __shared__ float d[CH];

    const int b = blockIdx.x;
    const int t = threadIdx.x;  // 0..63, acts as column index (row step) / row index (col step)

    const float* Sb = S_in + (size_t)b * CH * CH;
    for (int j = 0; j < CH; ++j) {
        S[t][j] = Sb[t * CH + j];
        U[t][j] = (t == j) ? 1.0f : 0.0f;
    }
    __syncthreads();

    for (int sweep = 0; sweep < 10; ++sweep) {
        for (int r = 0; r < 63; ++r) {
            if (t < 32) {
                int p, q;
                if (t == 0) { p = 63; q = r % 63; }
                else        { p = (r + t) % 63; q = (r + 63 - t) % 63; }
                if (p > q) { int tmp = p; p = q; q = tmp; }
                float app = S[p][p], aqq = S[q][q], apq = S[p][q];
                float c = 1.0f, s = 0.0f;
                if (fabsf(apq) > 1e-12f) {
                    float tau = (aqq - app) / (2.0f * apq);
                    float tt  = copysignf(1.0f, tau) /
                                (fabsf(tau) + sqrtf(1.0f + tau * tau));
                    c = rsqrtf(1.0f + tt * tt);
                    s = tt * c;
                }
                pp[t] = p; qq[t] = q; rc[t] = c; rs[t] = s;
            }
            __syncthreads();
            // Row update: thread t owns column t; S' = J^T S
            #pragma unroll 4
            for (int k = 0; k < 32; ++k) {
                const int p = pp[k], q = qq[k];
                const float c = rc[k], s = rs[k];
                const float sp = S[p][t], sq = S[q][t];
                S[p][t] = c * sp - s * sq;
                S[q][t] = s * sp + c * sq;
            }
            __syncthreads();
            // Column update: thread t owns row t; S'' = S' J, U' = U J
            #pragma unroll 4
            for (int k = 0; k < 32; ++k) {
                const int p = pp[k], q = qq[k];
                const float c = rc[k], s = rs[k];
                const float sp = S[t][p], sq = S[t][q];
                S[t][p] = c * sp - s * sq;
                S[t][q] = s * sp + c * sq;
                const float up = U[t][p], uq = U[t][q];
                U[t][p] = c * up - s * uq;
                U[t][q] = s * up + c * uq;
            }
            __syncthreads();
        }
    }

    d[t] = S[t][t];
    __syncthreads();
    const float dv = d[t];
    int rank = 0;
    for (int j = 0; j < CH; ++j) {
        const float dj = d[j];
        rank += (dj < dv) || (dj == dv && j < t);
    }
    float* Ub = U_out + (size_t)b * CH * CH;
    for (int i = 0; i < CH; ++i)
        Ub[i * CH + rank] = U[i][t];
}

// ---------------------------------------------------------------------------
// Kernel 3: y_rot[b] = U[b] @ y[b]   (64x64 x 64x4096), fp32 WMMA 16x16x4.
// Grid (8 N-chunks, 256 batches), 8 waves/block. Each wave produces a 64x16
// output strip per iteration (4 row-tiles accumulated over K=64).
// ---------------------------------------------------------------------------
__global__ __launch_bounds__(256) void rot_kernel(const float* __restrict__ U,
                                                  const float* __restrict__ y,
                                                  float* __restrict__ out) {
    const int b    = blockIdx.y;
    const int nblk = blockIdx.x;          // 512-column chunk
    const int wave = threadIdx.x >> 5;
    const int lane = threadIdx.x & 31;
    const int m    = lane & 15;
    const int h    = lane >> 4;

    const float* Ub = U + (size_t)b * CH * CH;
    const float* yb = y + (size_t)b * CH * NN;
    float* ob = out + (size_t)b * CH * NN;

    for (int nit = 0; nit < 4; ++nit) {
        const int ncol0 = nblk * 512 + (wave * 4 + nit) * 16;
        v8f acc[4];
        #pragma unroll
        for (int mi = 0; mi < 4; ++mi) acc[mi] = v8f{};

        const float* bcol = yb + ncol0 + m;  // element j: yb[(kk+j+2h)*NN + ncol0+m]
        #pragma unroll 4
        for (int kk = 0; kk < CH; kk += 4) {
            v2f bf;
            bf.x = bcol[(size_t)(kk + 2 * h)     * NN];
            bf.y = bcol[(size_t)(kk + 2 * h + 1) * NN];
            #pragma unroll
            for (int mi = 0; mi < 4; ++mi) {
                v2f af = *(const v2f*)(Ub + (mi * 16 + m) * CH + kk + 2 * h);
                acc[mi] = __builtin_amdgcn_wmma_f32_16x16x4_f32(
                    false, af, false, bf, (short)0, acc[mi], false, false);
            }
        }

        #pragma unroll
        for (int mi = 0; mi < 4; ++mi) {
            #pragma unroll
            for (int r = 0; r < 8; ++r) {
                const int row = mi * 16 + r + 8 * h;
                ob[(size_t)row * NN + ncol0 + m] = acc[mi][r];
            }
        }
    }
}

// ---------------------------------------------------------------------------
extern "C" void kernel_launch(void* const* d_in, const int* in_sizes, int n_in,
                              void* d_out, int out_size, void* d_ws, size_t ws_size,
                              hipStream_t stream) {
    (void)in_sizes; (void)n_in; (void)out_size; (void)ws_size;

    const float* y = (const float*)d_in[0];
    float* yrot = (float*)d_out;                             // BATCH*CH*NN floats
    float* Uout = (float*)d_out + (size_t)BATCH * CH * NN;   // BATCH*CH*CH floats
    float* S    = (float*)d_ws;                              // BATCH*CH*CH floats (4 MB)

    cov_kernel<<<dim3(BATCH), dim3(256), 0, stream>>>(y, S);
    jacobi_kernel<<<dim3(BATCH), dim3(64), 0, stream>>>(S, Uout);
    rot_kernel<<<dim3(8, BATCH), dim3(256), 0, stream>>>(Uout, y, yrot);
}